// SWA_10934986735777
// MI455X (gfx1250) — compile-verified
//
#include <hip/hip_runtime.h>

typedef __attribute__((ext_vector_type(2))) float v2f;
typedef __attribute__((ext_vector_type(8))) float v8f;

constexpr int BB     = 2;             // batch
constexpr int C      = 256;           // channels
constexpr int L      = 1024;          // sequence length
constexpr int H      = 8;             // heads
constexpr int D      = 32;            // head dim
constexpr int PADW   = 3;             // WINDOW // 2
constexpr int P      = L + 2 * PADW;  // 1030 real padded positions
constexpr int PPAD   = 1040;          // K/V position stride
constexpr int XTROWS = 1056;          // xT rows (33 tiles of 32), zero-padded

// Multiplicity of padded position p in the flattened (Wd, L) key axis.
__device__ __forceinline__ float cntf(int p) {
    int lo = p < 6 ? p : 6;
    int hi = p - (L - 1); if (hi < 0) hi = 0;
    return (float)(lo - hi + 1);
}

// ---------------------------------------------------------------------------
// Stage 0: zero-padded transpose  x[B][C][L] -> xT[B][XTROWS][C]
//   xT[b][row][c] = x[b][c][row - PADW]  (0 outside [0, L))
// ---------------------------------------------------------------------------
__global__ void __launch_bounds__(256)
swa_transpose_pad(const float* __restrict__ x, float* __restrict__ xT)
{
    __shared__ float tile[32][33];
    const int b  = blockIdx.z;
    const int c0 = blockIdx.y * 32;
    const int p0 = blockIdx.x * 32;
    const int tx = threadIdx.x;   // 0..31
    const int ty = threadIdx.y;   // 0..7

    #pragma unroll
    for (int i = 0; i < 4; ++i) {
        const int c = c0 + ty * 4 + i;
        const int l = p0 + tx - PADW;
        tile[ty * 4 + i][tx] =
            (l >= 0 && l < L) ? x[((size_t)b * C + c) * L + l] : 0.0f;
    }
    __syncthreads();
    #pragma unroll
    for (int i = 0; i < 4; ++i) {
        const int p = p0 + ty * 4 + i;
        xT[((size_t)b * XTROWS + p) * C + c0 + tx] = tile[tx][ty * 4 + i];
    }
}

// ---------------------------------------------------------------------------
// Pointwise-conv projection as WMMA GEMM on the transposed input:
//   out[o, pos] = W[o,:] . xT[row(pos), :] + bias[o]
// One wave -> 16 x 32 output tile (two accumulators sharing each A fragment).
// layout 0 (Q): row = pos + PADW, out [B][H][L][D]
// layout 1 (K): row = pos,        out [B][H][PPAD][D]
// layout 2 (V): row = pos,        out [B][H][D][PPAD]  (transposed for O=V^T.P^T)
// ---------------------------------------------------------------------------
__global__ void __launch_bounds__(32)
swa_proj_wmma(const float* __restrict__ xT, const float* __restrict__ W,
              const float* __restrict__ bias, float* __restrict__ out, int layout)
{
    const int lane  = threadIdx.x & 31;
    const int col16 = lane & 15;
    const int khalf = (lane >> 4) << 1;   // 0 or 2 (A/B fragment K pair)
    const int rbase = (lane >> 4) << 3;   // 0 or 8 (C/D fragment row base)

    const int m0 = blockIdx.x * 16;                 // output-channel tile
    const int tiles_per_b = layout ? 33 : 32;       // 32-col tiles
    const int b  = blockIdx.y / tiles_per_b;
    const int n0 = (blockIdx.y % tiles_per_b) * 32;

    const int pos0 = n0 + col16;
    const int pos1 = pos0 + 16;
    const int row0 = pos0 + (layout ? 0 : PADW);

    const float* wrow = W  + (size_t)(m0 + col16) * C + khalf;
    const float* xr0  = xT + ((size_t)b * XTROWS + row0) * C + khalf;
    const float* xr1  = xr0 + (size_t)16 * C;

    v8f acc0 = {}, acc1 = {};
    for (int kk = 0; kk < C / 4; ++kk) {
        const int k = kk * 4;
        const v2f a  = *(const v2f*)(wrow + k);
        const v2f b0 = *(const v2f*)(xr0 + k);
        const v2f b1 = *(const v2f*)(xr1 + k);
        acc0 = __builtin_amdgcn_wmma_f32_16x16x4_f32(false, a, false, b0,
                                                     (short)0, acc0, false, false);
        acc1 = __builtin_amdgcn_wmma_f32_16x16x4_f32(false, a, false, b1,
                                                     (short)0, acc1, false, false);
    }

    #pragma unroll
    for (int r = 0; r < 8; ++r) {
        const int o  = m0 + r + rbase;   // output channel
        const int h  = o >> 5;
        const int dd = o & 31;
        const float bv = bias[o];
        if (layout == 2) {               // V: [B][H][D][PPAD]
            float* vb = out + (((size_t)b * H + h) * D + dd) * PPAD;
            vb[pos0] = acc0[r] + bv;
            if (pos1 < PPAD) vb[pos1] = acc1[r] + bv;
        } else {                         // Q/K: [B][H][posdim][D]
            const int posdim = layout ? PPAD : L;
            float* qb = out + (((size_t)b * H + h) * posdim) * D + dd;
            qb[(size_t)pos0 * D] = acc0[r] + bv;
            if (!layout || pos1 < PPAD) qb[(size_t)pos1 * D] = acc1[r] + bv;
        }
    }
}

// ---------------------------------------------------------------------------
// Single-pass flash attention for one (b, h, 16-query tile).
// Per 16-key tile: S = Q.K^T (8 WMMA), online count-weighted softmax stats,
// rescale O^T accumulators via per-column factor bounced through LDS, then
// O^T += V^T . P^T (8 WMMA, P transposed through LDS). Final 1/Z + gamma*O+x.
// ---------------------------------------------------------------------------
__global__ void __launch_bounds__(32)
swa_attn_wmma(const float* __restrict__ Q, const float* __restrict__ K,
              const float* __restrict__ V, const float* __restrict__ x,
              const float* __restrict__ gamma, float* __restrict__ out)
{
    __shared__ float ldsP[16 * 16];   // transposed (unnormalized) P tile
    __shared__ float ldsF[16];        // per-query rescale factor
    __shared__ float ldsZ[16];        // per-query denominator

    const int lane  = threadIdx.x & 31;
    const int col16 = lane & 15;
    const int khalf = (lane >> 4) << 1;
    const int rbase = (lane >> 4) << 3;

    const int l0 = blockIdx.x * 16;
    const int h  = blockIdx.y;
    const int b  = blockIdx.z;

    const float* Qb = Q + (((size_t)b * H + h) * L) * D;
    const float* Kb = K + (((size_t)b * H + h) * PPAD) * D;
    const float* Vb = V + (((size_t)b * H + h) * D) * PPAD;

    // Q A-fragments: row = l0+col16, K dim = d (32) in 8 steps of 4
    v2f qa[8];
    #pragma unroll
    for (int kk = 0; kk < 8; ++kk)
        qa[kk] = *(const v2f*)(Qb + (size_t)(l0 + col16) * D + kk * 4 + khalf);

    float mrow[8], zrow[8];
    #pragma unroll
    for (int r = 0; r < 8; ++r) { mrow[r] = -3.0e38f; zrow[r] = 0.0f; }

    v8f o0 = {}, o1 = {};
    const int NT = (P + 15) / 16;   // 65 key tiles

    for (int t = 0; t < NT; ++t) {
        const int  p0 = t * 16;
        const int  p  = p0 + col16;       // < PPAD: always a valid load
        const bool pv = p < P;

        if (p + 16 < PPAD)
            __builtin_prefetch(Kb + (size_t)(p + 16) * D, 0, 0);

        // ---- S = Q . K^T ----
        v8f s = {};
        #pragma unroll
        for (int kk = 0; kk < 8; ++kk) {
            const v2f kf = *(const v2f*)(Kb + (size_t)p * D + kk * 4 + khalf);
            s = __builtin_amdgcn_wmma_f32_16x16x4_f32(false, qa[kk], false, kf,
                                                      (short)0, s, false, false);
        }

        // ---- online softmax stats + unnormalized P to LDS (transposed) ----
        const float cw = pv ? cntf(p) : 0.0f;
        #pragma unroll
        for (int r = 0; r < 8; ++r) {
            float sv = pv ? s[r] : -3.0e38f;
            float tm = sv;                          // reduce over 16 lanes of
            tm = fmaxf(tm, __shfl_xor(tm, 1, 32));  // this half (same S rows)
            tm = fmaxf(tm, __shfl_xor(tm, 2, 32));
            tm = fmaxf(tm, __shfl_xor(tm, 4, 32));
            tm = fmaxf(tm, __shfl_xor(tm, 8, 32));
            const float nm = fmaxf(mrow[r], tm);
            const float pe = cw * __expf(sv - nm);  // 0 for masked columns
            float e = pe;
            e += __shfl_xor(e, 1, 32);
            e += __shfl_xor(e, 2, 32);
            e += __shfl_xor(e, 4, 32);
            e += __shfl_xor(e, 8, 32);
            const float f = __expf(mrow[r] - nm);   // accumulator rescale
            zrow[r] = zrow[r] * f + e;
            mrow[r] = nm;
            ldsP[col16 * 16 + rbase + r] = pe;      // P^T[p_local][l_local]
            if (col16 == r) ldsF[rbase + r] = f;    // one writer per row
        }
        __syncthreads();

        // ---- rescale O^T (columns are queries) and accumulate V^T . P^T ----
        const float fc = ldsF[col16];
        #pragma unroll
        for (int r = 0; r < 8; ++r) { o0[r] *= fc; o1[r] *= fc; }

        #pragma unroll
        for (int kk = 0; kk < 4; ++kk) {
            v2f pb;                      // B frag: P^T[k=p_local][n=l_local]
            pb.x = ldsP[(kk * 4 + khalf)     * 16 + col16];
            pb.y = ldsP[(kk * 4 + khalf + 1) * 16 + col16];
            const int pa = p0 + kk * 4 + khalf;       // pa+1 <= 1039 < PPAD
            const v2f va0 = *(const v2f*)(Vb + (size_t)col16        * PPAD + pa);
            const v2f va1 = *(const v2f*)(Vb + (size_t)(col16 + 16) * PPAD + pa);
            o0 = __builtin_amdgcn_wmma_f32_16x16x4_f32(false, va0, false, pb,
                                                       (short)0, o0, false, false);
            o1 = __builtin_amdgcn_wmma_f32_16x16x4_f32(false, va1, false, pb,
                                                       (short)0, o1, false, false);
        }
        __syncthreads();
    }

    // ---- publish per-query denominators, normalize, epilogue g*O + x ----
    #pragma unroll
    for (int r = 0; r < 8; ++r)
        if (col16 == r) ldsZ[rbase + r] = zrow[r];
    __syncthreads();

    const float g  = gamma[0];
    const float gs = g / ldsZ[col16];   // column l0+col16's gamma/Z
    const int   ll = l0 + col16;
    #pragma unroll
    for (int r = 0; r < 8; ++r) {
        const int dd = r + rbase;
        const int c0 = h * D + dd;
        const size_t i0 = ((size_t)b * C + c0) * L + ll;
        out[i0] = gs * o0[r] + x[i0];
        const size_t i1 = ((size_t)b * C + c0 + 16) * L + ll;
        out[i1] = gs * o1[r] + x[i1];
    }
}

// ---------------------------------------------------------------------------
extern "C" void kernel_launch(void* const* d_in, const int* in_sizes, int n_in,
                              void* d_out, int out_size, void* d_ws, size_t ws_size,
                              hipStream_t stream)
{
    (void)in_sizes; (void)n_in; (void)out_size; (void)ws_size;

    const float* x     = (const float*)d_in[0];
    const float* Wq    = (const float*)d_in[1];
    const float* bq    = (const float*)d_in[2];
    const float* Wk    = (const float*)d_in[3];
    const float* bk    = (const float*)d_in[4];
    const float* Wv    = (const float*)d_in[5];
    const float* bv    = (const float*)d_in[6];
    const float* gamma = (const float*)d_in[7];
    float* out = (float*)d_out;

    float* xT  = (float*)d_ws;                              // [B][XTROWS][C]
    float* Qws = xT  + (size_t)BB * XTROWS * C;             // [B][H][L][D]
    float* Kws = Qws + (size_t)BB * H * L * D;              // [B][H][PPAD][D]
    float* Vws = Kws + (size_t)BB * H * PPAD * D;           // [B][H][D][PPAD]

    swa_transpose_pad<<<dim3(XTROWS / 32, C / 32, BB), dim3(32, 8), 0, stream>>>(x, xT);

    dim3 blk(32);
    swa_proj_wmma<<<dim3(C / 16, BB * 32), blk, 0, stream>>>(xT, Wq, bq, Qws, 0);
    swa_proj_wmma<<<dim3(C / 16, BB * 33), blk, 0, stream>>>(xT, Wk, bk, Kws, 1);
    swa_proj_wmma<<<dim3(C / 16, BB * 33), blk, 0, stream>>>(xT, Wv, bv, Vws, 2);
    swa_attn_wmma<<<dim3(L / 16, H, BB), blk, 0, stream>>>(Qws, Kws, Vws, x, gamma, out);
}